// PNN_75849122447426
// MI455X (gfx1250) — compile-verified
//
#include <hip/hip_runtime.h>
#include <hip/hip_bf16.h>
#include <math.h>

typedef __attribute__((ext_vector_type(2))) float        v2f;
typedef __attribute__((ext_vector_type(8))) float        v8f;
typedef __attribute__((ext_vector_type(4))) unsigned int v4u;
typedef __attribute__((ext_vector_type(8))) int          v8i;
typedef __attribute__((ext_vector_type(4))) int          v4i;

#define FIELD_NUM 26
#define EMBED_DIM 32
#define VOCAB     100000
#define BATCH     16384
#define K1TOT     832      // 26*32
#define KCAT      1508     // 832 + 676
#define KPAD      1536     // K padded to multiple of 32 (zero weights beyond 1508)
#define H0        256
#define H1        128
#define L1DIM     269
#define L1PAD     272      // padded to multiple of 16

// ---------------------------------------------------------------------------
// TDM 2D tile load: builds a D# per CDNA5 ISA 8.3/8.4 and issues
// tensor_load_to_lds (6-arg toolchain form: g0, g1, g2, g3, g4, cpol).
// data_size = 4 bytes. Optional LDS padding (stride pad) via
// pad_interval/pad_amount codes (interval code c => every 2^(c+1) DWORDs,
// amount code a => a+1 DWORDs inserted).
// ---------------------------------------------------------------------------
__device__ __forceinline__ void tdm_load_2d(
    const void* lds_ptr, const void* gptr,
    unsigned tensor_dim0, unsigned tensor_dim1,
    unsigned tile_dim0, unsigned tile_dim1,
    unsigned long long stride0_elems,
    int pad_enable, unsigned pad_interval_code, unsigned pad_amount_code)
{
    const unsigned long long ga = (unsigned long long)(uintptr_t)gptr;
    const unsigned lds_off = (unsigned)(uintptr_t)lds_ptr;   // low 32 bits = LDS byte addr

    v4u g0;
    g0[0] = 1u;                                          // count=1, is_restore=0
    g0[1] = lds_off;                                     // lds_addr [63:32]
    g0[2] = (unsigned)(ga & 0xFFFFFFFFull);              // global_addr[31:0]
    g0[3] = (unsigned)((ga >> 32) & 0x1FFFFFFull)        // global_addr[56:32]
          | (2u << 30);                                  // type = 2 (image)

    v8i g1;
    unsigned w0 = (2u << 16)                             // data_size = 4 bytes
                | ((unsigned)(pad_enable ? 1 : 0) << 20)
                | (pad_interval_code << 22)
                | (pad_amount_code << 25);               // workgroup_mask = 0
    g1[0] = (int)w0;
    g1[1] = (int)((tensor_dim0 & 0xFFFFu) << 16);                          // [63:48]=dim0 lo
    g1[2] = (int)((tensor_dim0 >> 16) | ((tensor_dim1 & 0xFFFFu) << 16));  // dim0 hi | dim1 lo
    g1[3] = (int)((tensor_dim1 >> 16) | (tile_dim0 << 16));                // dim1 hi | tile0
    g1[4] = (int)(tile_dim1 & 0xFFFFu);                                    // tile1 | tile2=0
    g1[5] = (int)(unsigned)(stride0_elems & 0xFFFFFFFFull);                // dim0_stride lo
    g1[6] = (int)(unsigned)((stride0_elems >> 32) & 0xFFFFull);            // dim0_stride hi
    g1[7] = 0;                                                             // dim1_stride (2D)

    v4i g2 = {0, 0, 0, 0};
    v4i g3 = {0, 0, 0, 0};
    v8i g4 = {0, 0, 0, 0, 0, 0, 0, 0};
    __builtin_amdgcn_tensor_load_to_lds(g0, g1, g2, g3, g4, 0);
}

// ---------------------------------------------------------------------------
// Kernel 1: embedding gather + pairwise interaction p = E * E^T via WMMA f32.
// One wave (32 lanes) per batch row. E (26x32, padded 32x32) staged in LDS.
// Writes X[b][0..831]  = embed flat,  X[b][832..1507] = p flat.
// ---------------------------------------------------------------------------
__global__ __launch_bounds__(256) void k_gather_interact(
    const float* __restrict__ tables, const int* __restrict__ sparse,
    float* __restrict__ X)
{
    __shared__ float E[8][32][33];                // 33: bank-conflict pad
    const int wave = threadIdx.x >> 5;
    const int lane = threadIdx.x & 31;
    const int b    = blockIdx.x * 8 + wave;

    for (int f = 0; f < 32; ++f) {
        float v = 0.0f;
        if (f < FIELD_NUM) {
            const int idx = sparse[b * FIELD_NUM + f];
            v = tables[(size_t)f * VOCAB * EMBED_DIM + (size_t)idx * EMBED_DIM + lane];
            X[(size_t)b * KPAD + f * EMBED_DIM + lane] = v;
        }
        E[wave][f][lane] = v;
    }
    __syncthreads();

    const int half = lane >> 4;
    const int l16  = lane & 15;
    v8f zero = {0.f,0.f,0.f,0.f,0.f,0.f,0.f,0.f};

    #pragma unroll
    for (int mi = 0; mi < 2; ++mi) {
        #pragma unroll
        for (int ni = 0; ni < 2; ++ni) {
            v8f acc = zero;
            #pragma unroll
            for (int k = 0; k < 8; ++k) {        // K = 32 = 8 steps of 4
                const int kc = k * 4 + half * 2;
                v2f a, bb;
                a.x  = E[wave][mi * 16 + l16][kc];
                a.y  = E[wave][mi * 16 + l16][kc + 1];
                bb.x = E[wave][ni * 16 + l16][kc];   // B = E^T (symmetric trick)
                bb.y = E[wave][ni * 16 + l16][kc + 1];
                acc = __builtin_amdgcn_wmma_f32_16x16x4_f32(
                        false, a, false, bb, (short)0, acc, false, false);
            }
            #pragma unroll
            for (int r = 0; r < 8; ++r) {
                const int m = mi * 16 + r + 8 * half;
                const int g = ni * 16 + l16;
                if (m < FIELD_NUM && g < FIELD_NUM)
                    X[(size_t)b * KPAD + K1TOT + m * FIELD_NUM + g] = acc[r];
            }
        }
    }
}

// ---------------------------------------------------------------------------
// Weight packing: Wcat = [w_z (832x256); w_p (676x256); zeros] -> 1536x256
// ---------------------------------------------------------------------------
__global__ void k_pack_wcat(const float* __restrict__ wz, const float* __restrict__ wp,
                            float* __restrict__ Wcat)
{
    const int t = blockIdx.x * blockDim.x + threadIdx.x;   // KPAD*H0 threads
    const int k = t >> 8, n = t & 255;
    float v = 0.0f;
    if (k < K1TOT)      v = wz[k * H0 + n];
    else if (k < KCAT)  v = wp[(k - K1TOT) * H0 + n];
    Wcat[t] = v;
}

__global__ void k_pack_w1(const float* __restrict__ W1, float* __restrict__ W1p)
{
    const int t = blockIdx.x * blockDim.x + threadIdx.x;   // L1PAD*H1 threads
    const int k = t >> 7, n = t & 127;
    W1p[t] = (k < L1DIM) ? W1[k * H1 + n] : 0.0f;
}

// dense part of concat: L1[b][256..271] = relu(dense[b][0..12]) | zeros
__global__ void k_dense_relu(const float* __restrict__ dense, float* __restrict__ L1)
{
    const int t = blockIdx.x * blockDim.x + threadIdx.x;   // BATCH*16 threads
    const int b = t >> 4, j = t & 15;
    float v = 0.0f;
    if (j < 13) { v = dense[b * 13 + j]; v = v > 0.0f ? v : 0.0f; }
    L1[(size_t)b * L1PAD + H0 + j] = v;
}

// ---------------------------------------------------------------------------
// Kernel 2: L1[:, 0:256] = relu( X(16384x1536) @ Wcat(1536x256) + l_b )
// Block = 256 thr (8 waves), tile M=128 N=64, K chunked by 32.
// Tiles staged into LDS by the Tensor Data Mover (double buffered);
// hardware pad gives As stride 36 / Bs stride 68 (bank-conflict free).
// ---------------------------------------------------------------------------
__global__ __launch_bounds__(256) void k_gemm1(
    const float* __restrict__ X, const float* __restrict__ Wcat,
    const float* __restrict__ lb, float* __restrict__ L1)
{
    __shared__ float As[2][128][36];     // 128x32 + TDM pad (interval 32, amount 4)
    __shared__ float Bs[2][32][68];      // 32x64  + TDM pad (interval 64, amount 4)
    const int tid  = threadIdx.x;
    const int wave = tid >> 5, lane = tid & 31;
    const int half = lane >> 4, l16 = lane & 15;
    const int ntile = blockIdx.x & 3;
    const size_t b0 = (size_t)(blockIdx.x >> 2) * 128;
    const int n0 = ntile * 64;

    v8f zero = {0.f,0.f,0.f,0.f,0.f,0.f,0.f,0.f};
    v8f acc[4] = { zero, zero, zero, zero };

    // preload K-chunk 0 into buffer 0
    if (wave == 0) {
        tdm_load_2d(&As[0][0][0], &X[b0 * KPAD], KPAD, BATCH, 32, 128,
                    KPAD, 1, 4, 3);                       // pad every 32 DW, +4 DW
        tdm_load_2d(&Bs[0][0][0], &Wcat[n0], H0, KPAD, 64, 32,
                    H0, 1, 5, 3);                         // pad every 64 DW, +4 DW
        __builtin_amdgcn_s_wait_tensorcnt(0);
    }
    __syncthreads();

    int buf = 0;
    for (int kb = 0; kb < KPAD; kb += 32) {
        // kick off DMA for the next K-chunk into the other buffer
        if (kb + 32 < KPAD && wave == 0) {
            tdm_load_2d(&As[buf ^ 1][0][0], &X[b0 * KPAD + kb + 32], KPAD, BATCH,
                        32, 128, KPAD, 1, 4, 3);
            tdm_load_2d(&Bs[buf ^ 1][0][0], &Wcat[(size_t)(kb + 32) * H0 + n0], H0, KPAD,
                        64, 32, H0, 1, 5, 3);
        }

        #pragma unroll
        for (int kk = 0; kk < 8; ++kk) {
            const int kc = kk * 4 + half * 2;
            v2f a;
            a.x = As[buf][wave * 16 + l16][kc];
            a.y = As[buf][wave * 16 + l16][kc + 1];
            #pragma unroll
            for (int ni = 0; ni < 4; ++ni) {
                v2f bb;
                bb.x = Bs[buf][kc][ni * 16 + l16];
                bb.y = Bs[buf][kc + 1][ni * 16 + l16];
                acc[ni] = __builtin_amdgcn_wmma_f32_16x16x4_f32(
                            false, a, false, bb, (short)0, acc[ni], false, false);
            }
        }

        if (wave == 0) __builtin_amdgcn_s_wait_tensorcnt(0);
        __syncthreads();
        buf ^= 1;
    }

    #pragma unroll
    for (int ni = 0; ni < 4; ++ni) {
        const int n = n0 + ni * 16 + l16;
        const float bias = lb[n];
        #pragma unroll
        for (int r = 0; r < 8; ++r) {
            const size_t row = b0 + wave * 16 + r + 8 * half;
            float v = acc[ni][r] + bias;
            v = v > 0.0f ? v : 0.0f;
            L1[row * L1PAD + n] = v;
        }
    }
}

// ---------------------------------------------------------------------------
// Kernel 3: h = relu(L1(16384x272) @ W1p(272x128) + b1);
//           out = sigmoid(h @ Wf + bf)   (fused epilogue, shfl reduction)
// Block = 256 thr (8 waves), tile M=128 N=128, K chunked by 16, TDM staged.
// ---------------------------------------------------------------------------
__global__ __launch_bounds__(256) void k_gemm2_out(
    const float* __restrict__ L1, const float* __restrict__ W1p,
    const float* __restrict__ b1, const float* __restrict__ Wf,
    const float* __restrict__ bfp, float* __restrict__ out)
{
    __shared__ float As[2][128][20];     // 128x16 + TDM pad (interval 16, amount 4)
    __shared__ float Bs[2][16][132];     // 16x128 + TDM pad (interval 128, amount 4)
    const int tid  = threadIdx.x;
    const int wave = tid >> 5, lane = tid & 31;
    const int half = lane >> 4, l16 = lane & 15;
    const size_t b0 = (size_t)blockIdx.x * 128;

    v8f zero = {0.f,0.f,0.f,0.f,0.f,0.f,0.f,0.f};
    v8f acc[8] = { zero, zero, zero, zero, zero, zero, zero, zero };

    if (wave == 0) {
        tdm_load_2d(&As[0][0][0], &L1[b0 * L1PAD], L1PAD, BATCH, 16, 128,
                    L1PAD, 1, 3, 3);                      // pad every 16 DW, +4 DW
        tdm_load_2d(&Bs[0][0][0], &W1p[0], H1, L1PAD, 128, 16,
                    H1, 1, 6, 3);                         // pad every 128 DW, +4 DW
        __builtin_amdgcn_s_wait_tensorcnt(0);
    }
    __syncthreads();

    int buf = 0;
    for (int kb = 0; kb < L1PAD; kb += 16) {
        if (kb + 16 < L1PAD && wave == 0) {
            tdm_load_2d(&As[buf ^ 1][0][0], &L1[b0 * L1PAD + kb + 16], L1PAD, BATCH,
                        16, 128, L1PAD, 1, 3, 3);
            tdm_load_2d(&Bs[buf ^ 1][0][0], &W1p[(size_t)(kb + 16) * H1], H1, L1PAD,
                        128, 16, H1, 1, 6, 3);
        }

        #pragma unroll
        for (int kk = 0; kk < 4; ++kk) {
            const int kc = kk * 4 + half * 2;
            v2f a;
            a.x = As[buf][wave * 16 + l16][kc];
            a.y = As[buf][wave * 16 + l16][kc + 1];
            #pragma unroll
            for (int ni = 0; ni < 8; ++ni) {
                v2f bb;
                bb.x = Bs[buf][kc][ni * 16 + l16];
                bb.y = Bs[buf][kc + 1][ni * 16 + l16];
                acc[ni] = __builtin_amdgcn_wmma_f32_16x16x4_f32(
                            false, a, false, bb, (short)0, acc[ni], false, false);
            }
        }

        if (wave == 0) __builtin_amdgcn_s_wait_tensorcnt(0);
        __syncthreads();
        buf ^= 1;
    }

    // fused: relu(+b1), dot with Wf over N=128, sigmoid(+bf)
    float psum[8];
    #pragma unroll
    for (int r = 0; r < 8; ++r) psum[r] = 0.0f;
    #pragma unroll
    for (int ni = 0; ni < 8; ++ni) {
        const int n = ni * 16 + l16;
        const float bias = b1[n];
        const float wf   = Wf[n];
        #pragma unroll
        for (int r = 0; r < 8; ++r) {
            float h = acc[ni][r] + bias;
            h = h > 0.0f ? h : 0.0f;
            psum[r] += h * wf;
        }
    }
    const float bfv = bfp[0];
    #pragma unroll
    for (int r = 0; r < 8; ++r) {
        float s = psum[r];
        s += __shfl_xor(s, 1, 32);
        s += __shfl_xor(s, 2, 32);
        s += __shfl_xor(s, 4, 32);
        s += __shfl_xor(s, 8, 32);     // reduce over the 16 lanes of each half
        if (l16 == 0) {
            const size_t row = b0 + wave * 16 + r + 8 * half;
            const float z = s + bfv;
            out[row] = 1.0f / (1.0f + __expf(-z));
        }
    }
}

// ---------------------------------------------------------------------------
extern "C" void kernel_launch(void* const* d_in, const int* in_sizes, int n_in,
                              void* d_out, int out_size, void* d_ws, size_t ws_size,
                              hipStream_t stream)
{
    const float* dense  = (const float*)d_in[0];
    const int*   sparse = (const int*)  d_in[1];
    const float* tables = (const float*)d_in[2];
    const float* wz     = (const float*)d_in[3];
    const float* wp     = (const float*)d_in[4];
    const float* lb     = (const float*)d_in[5];
    const float* W1     = (const float*)d_in[6];
    const float* b1     = (const float*)d_in[7];
    const float* Wf     = (const float*)d_in[8];
    const float* bf     = (const float*)d_in[9];
    float* out = (float*)d_out;

    float* ws   = (float*)d_ws;
    float* X    = ws;                                   // BATCH*KPAD
    float* L1   = X    + (size_t)BATCH * KPAD;          // BATCH*L1PAD
    float* Wcat = L1   + (size_t)BATCH * L1PAD;         // KPAD*H0
    float* W1p  = Wcat + (size_t)KPAD * H0;             // L1PAD*H1

    k_pack_wcat     <<<(KPAD * H0) / 256,  256, 0, stream>>>(wz, wp, Wcat);
    k_pack_w1       <<<(L1PAD * H1) / 256, 256, 0, stream>>>(W1, W1p);
    k_dense_relu    <<<(BATCH * 16) / 256, 256, 0, stream>>>(dense, L1);
    k_gather_interact<<<BATCH / 8,         256, 0, stream>>>(tables, sparse, X);
    k_gemm1         <<<(BATCH / 128) * (H0 / 64), 256, 0, stream>>>(X, Wcat, lb, L1);
    k_gemm2_out     <<<BATCH / 128,        256, 0, stream>>>(L1, W1p, b1, Wf, bf, out);
}